// Model_55757265437245
// MI455X (gfx1250) — compile-verified
//
#include <hip/hip_runtime.h>

#define N_DRUG 50000
#define N_GENE 50000
#define NEDGE  600000
#define D_IN   128
#define D_HID  64
#define D_EMB  32

typedef __attribute__((ext_vector_type(2))) float v2f;
typedef __attribute__((ext_vector_type(8))) float v8f;

// ---------------- utility kernels ----------------

__global__ void fill_zero(float* __restrict__ p, long n) {
    long i = (long)blockIdx.x * blockDim.x + threadIdx.x;
    if (i < n) p[i] = 0.0f;
}

__global__ void init_out_bias(float* __restrict__ out,
                              const float* __restrict__ bd,
                              const float* __restrict__ bg) {
    int i = blockIdx.x * blockDim.x + threadIdx.x;
    const int total = (N_DRUG + N_GENE) * D_EMB;
    if (i >= total) return;
    int row = i >> 5;           // D_EMB == 32
    int f   = i & (D_EMB - 1);
    out[i] = (row < N_DRUG) ? bd[f] : bg[f];
}

__global__ void degree_kernel(const int* __restrict__ dst,
                              float* __restrict__ deg, int E) {
    int e = blockIdx.x * blockDim.x + threadIdx.x;
    if (e < E) atomicAdd(&deg[dst[e]], 1.0f);
}

__global__ void invdeg_kernel(float* __restrict__ deg, long n) {
    long i = (long)blockIdx.x * blockDim.x + threadIdx.x;
    if (i < n) deg[i] = 1.0f / fmaxf(deg[i], 1.0f);
}

__global__ void bias_relu_kernel(float* __restrict__ h,
                                 const float* __restrict__ b, long n) {
    long i = (long)blockIdx.x * blockDim.x + threadIdx.x;
    if (i < n) {
        float v = h[i] + b[i & (D_HID - 1)];   // D_HID == 64
        h[i] = v > 0.0f ? v : 0.0f;
    }
}

// ---------------- fp32 WMMA GEMM: M[N x O] = X[N x K] @ W[K x O] ----------------
// One wave owns one column tile (tn) and sweeps a strip of MPW M-tiles.
// All K/4 B fragments are preloaded into registers once (W is tiny / L2-hot),
// so the steady state is: one float2 A load per V_WMMA_F32_16X16X4_F32.
// Two accumulators (even/odd k-steps) break the WMMA RAW chain for ILP.
//
// fp32 A 16x4 layout (ISA 7.12.2): lane<16 -> row=lane, K pair {0,1};
// lane>=16 -> row=lane-16, K pair {2,3}. B (4x16) symmetric (col=lane&15).
// C/D: VGPR i -> row i (lanes 0-15) / row i+8 (lanes 16-31), col = lane&15.
template<int K, int O, int MPW>
__global__ void gemm_wmma_f32(const float* __restrict__ X,
                              const float* __restrict__ W,
                              float* __restrict__ M,
                              int tilesM, int totalWaves) {
    constexpr int KK     = K / 4;     // WMMA k-steps (even: 32 or 16)
    constexpr int tilesN = O / 16;

    int wid = blockIdx.x * (blockDim.x >> 5) + (threadIdx.x >> 5);
    if (wid >= totalWaves) return;    // whole-wave exit: EXEC stays all-ones

    int strip = wid / tilesN;
    int tn    = wid - strip * tilesN;
    int tm0   = strip * MPW;

    int lane = threadIdx.x & 31;
    int r    = lane & 15;             // row of A / col of B,C
    int kh   = lane >> 4;             // selects K pair {0,1} vs {2,3}

    // preload all B fragments for this column tile
    v2f bfrag[KK];
    const float* __restrict__ wcol = W + (long)(2 * kh) * O + tn * 16 + r;
#pragma unroll
    for (int kk = 0; kk < KK; ++kk) {
        bfrag[kk][0] = wcol[(long)(4 * kk) * O];
        bfrag[kk][1] = wcol[(long)(4 * kk + 1) * O];
    }

    for (int t = 0; t < MPW; ++t) {
        int tm = tm0 + t;
        if (tm >= tilesM) break;      // wave-uniform branch
        const float* __restrict__ xrow = X + (long)(tm * 16 + r) * K + 2 * kh;

        if (t + 1 < MPW && tm + 1 < tilesM)
            __builtin_prefetch(xrow + 16 * K, 0, 1);   // global_prefetch_b8

        v8f c0 = {}, c1 = {};
#pragma unroll
        for (int kk = 0; kk < KK; kk += 2) {
            float2 a0v = *(const float2*)(xrow + 4 * kk);
            float2 a1v = *(const float2*)(xrow + 4 * (kk + 1));
            v2f a0; a0[0] = a0v.x; a0[1] = a0v.y;
            v2f a1; a1[0] = a1v.x; a1[1] = a1v.y;
            c0 = __builtin_amdgcn_wmma_f32_16x16x4_f32(
                     false, a0, false, bfrag[kk],     (short)0, c0, false, false);
            c1 = __builtin_amdgcn_wmma_f32_16x16x4_f32(
                     false, a1, false, bfrag[kk + 1], (short)0, c1, false, false);
        }
        v8f c = c0 + c1;

        float* __restrict__ orow = M + (long)(tm * 16 + 8 * kh) * O + tn * 16 + r;
#pragma unroll
        for (int i = 0; i < 8; ++i) orow[(long)i * O] = c[i];
    }
}

// ---------------- edge aggregation: acc[dst] += m[src] * invdeg[dst] ----------------
// One thread handles 4 floats of one edge message (float4 load, 4 float atomics).
// m and acc tables are ~12.8 MB each -> L2 (192 MB) resident; atomics land in L2.
__global__ void edge_agg(const float* __restrict__ m,
                         const int* __restrict__ src,
                         const int* __restrict__ dst,
                         const float* __restrict__ invdeg,
                         float* __restrict__ acc,
                         int qbits /* log2(O/4) */, int O) {
    long tid = (long)blockIdx.x * blockDim.x + threadIdx.x;
    long total = (long)NEDGE << qbits;
    if (tid >= total) return;
    long e = tid >> qbits;
    int  q = (int)(tid & ((1 << qbits) - 1));

    int s = src[e];
    int d = dst[e];
    float w = invdeg[d];

    float4 v = ((const float4*)(m + (long)s * O))[q];
    float* o = acc + (long)d * O + q * 4;
    atomicAdd(o + 0, v.x * w);
    atomicAdd(o + 1, v.y * w);
    atomicAdd(o + 2, v.z * w);
    atomicAdd(o + 3, v.w * w);
}

// ---------------- host ----------------

extern "C" void kernel_launch(void* const* d_in, const int* in_sizes, int n_in,
                              void* d_out, int out_size, void* d_ws, size_t ws_size,
                              hipStream_t stream) {
    const float* x_drug = (const float*)d_in[0];
    const float* x_gene = (const float*)d_in[1];
    const int* dd_src = (const int*)d_in[2];
    const int* dd_dst = (const int*)d_in[3];
    const int* dg_src = (const int*)d_in[4];
    const int* dg_dst = (const int*)d_in[5];
    const int* gd_src = (const int*)d_in[6];
    const int* gd_dst = (const int*)d_in[7];
    const int* gg_src = (const int*)d_in[8];
    const int* gg_dst = (const int*)d_in[9];
    const float* W1_dd = (const float*)d_in[10];
    const float* W1_dg = (const float*)d_in[11];
    const float* W1_gd = (const float*)d_in[12];
    const float* W1_gg = (const float*)d_in[13];
    const float* b1_d  = (const float*)d_in[14];
    const float* b1_g  = (const float*)d_in[15];
    const float* W2_dd = (const float*)d_in[16];
    const float* W2_dg = (const float*)d_in[17];
    const float* W2_gd = (const float*)d_in[18];
    const float* W2_gg = (const float*)d_in[19];
    const float* b2_d  = (const float*)d_in[20];
    const float* b2_g  = (const float*)d_in[21];

    float* ws = (float*)d_ws;
    const long NPAD = 50176;                       // 50000 padded for alignment
    float* deg_dd = ws;
    float* deg_gd = ws + NPAD;
    float* deg_dg = ws + 2 * NPAD;
    float* deg_gg = ws + 3 * NPAD;
    float* accD   = ws + 4 * NPAD;                 // N_DRUG x 64, becomes hd
    float* accG   = accD + (long)N_DRUG * D_HID;   // N_GENE x 64, becomes hg
    float* mbuf   = accG + (long)N_GENE * D_HID;   // scratch for transformed feats

    // zero degrees + layer-1 accumulators (contiguous region)
    long zn = 4 * NPAD + (long)(N_DRUG + N_GENE) * D_HID;
    fill_zero<<<(int)((zn + 255) / 256), 256, 0, stream>>>(ws, zn);

    // per-relation degrees -> inverse degrees
    int eb = (NEDGE + 255) / 256;
    degree_kernel<<<eb, 256, 0, stream>>>(dd_dst, deg_dd, NEDGE);
    degree_kernel<<<eb, 256, 0, stream>>>(gd_dst, deg_gd, NEDGE);
    degree_kernel<<<eb, 256, 0, stream>>>(dg_dst, deg_dg, NEDGE);
    degree_kernel<<<eb, 256, 0, stream>>>(gg_dst, deg_gg, NEDGE);
    invdeg_kernel<<<(int)((4 * NPAD + 255) / 256), 256, 0, stream>>>(ws, 4 * NPAD);

    const int tilesM = N_DRUG / 16;                // 3125 = 5^5, exact
    const int MPW    = 5;                          // M-tiles per wave (divides 3125)
    const int strips = tilesM / MPW;               // 625

    auto gemm1 = [&](const float* X, const float* W, float* M) {
        int waves = strips * (D_HID / 16);
        gemm_wmma_f32<D_IN, D_HID, MPW><<<(waves + 7) / 8, 256, 0, stream>>>(
            X, W, M, tilesM, waves);
    };
    auto gemm2 = [&](const float* X, const float* W, float* M) {
        int waves = strips * (D_EMB / 16);
        gemm_wmma_f32<D_HID, D_EMB, MPW><<<(waves + 7) / 8, 256, 0, stream>>>(
            X, W, M, tilesM, waves);
    };
    auto agg = [&](const float* m, const int* s, const int* d, const float* inv,
                   float* acc, int O, int qbits) {
        long total = (long)NEDGE << qbits;
        edge_agg<<<(int)((total + 255) / 256), 256, 0, stream>>>(m, s, d, inv, acc, qbits, O);
    };

    // ---- layer 1 (K=128 -> O=64), mbuf reused serially on the stream ----
    gemm1(x_drug, W1_dd, mbuf); agg(mbuf, dd_src, dd_dst, deg_dd, accD, D_HID, 4);
    gemm1(x_gene, W1_gd, mbuf); agg(mbuf, gd_src, gd_dst, deg_gd, accD, D_HID, 4);
    gemm1(x_drug, W1_dg, mbuf); agg(mbuf, dg_src, dg_dst, deg_dg, accG, D_HID, 4);
    gemm1(x_gene, W1_gg, mbuf); agg(mbuf, gg_src, gg_dst, deg_gg, accG, D_HID, 4);

    long hn = (long)N_DRUG * D_HID;
    bias_relu_kernel<<<(int)((hn + 255) / 256), 256, 0, stream>>>(accD, b1_d, hn);
    bias_relu_kernel<<<(int)((hn + 255) / 256), 256, 0, stream>>>(accG, b1_g, hn);

    // ---- layer 2 (K=64 -> O=32), aggregate directly into d_out ----
    float* out = (float*)d_out;
    int ob = ((N_DRUG + N_GENE) * D_EMB + 255) / 256;
    init_out_bias<<<ob, 256, 0, stream>>>(out, b2_d, b2_g);

    float* out_gene = out + (long)N_DRUG * D_EMB;
    gemm2(accD, W2_dd, mbuf); agg(mbuf, dd_src, dd_dst, deg_dd, out,      D_EMB, 3);
    gemm2(accG, W2_gd, mbuf); agg(mbuf, gd_src, gd_dst, deg_gd, out,      D_EMB, 3);
    gemm2(accD, W2_dg, mbuf); agg(mbuf, dg_src, dg_dst, deg_dg, out_gene, D_EMB, 3);
    gemm2(accG, W2_gg, mbuf); agg(mbuf, gg_src, gg_dst, deg_gg, out_gene, D_EMB, 3);
}